// Aggregator_5549097746703
// MI455X (gfx1250) — compile-verified
//
#include <hip/hip_runtime.h>

#define N_ENT      100000
#define N_GENES    10000
#define N_FACTORS  4
#define N_REL      32
#define CH         64
#define N_KG_EDGES 2000000
#define N_SL_EDGES 160000
#define NNZ        200000

typedef __attribute__((ext_vector_type(2))) float v2f;
typedef __attribute__((ext_vector_type(8))) float v8f;

// ---------------- utility ----------------

__global__ void k_zero(float* __restrict__ p, size_t n) {
  size_t i = (size_t)blockIdx.x * blockDim.x + threadIdx.x;
  if (i < n) p[i] = 0.0f;
}

__global__ void k_init_minf(float* __restrict__ p, int n) {
  int i = blockIdx.x * blockDim.x + threadIdx.x;
  if (i < n) p[i] = -__builtin_inff();
}

// ordered float atomic-max via int bit tricks (init must be -inf)
__device__ __forceinline__ void atomicMaxF(float* addr, float val) {
  if (val >= 0.0f) atomicMax((int*)addr, __float_as_int(val));
  else             atomicMin((unsigned int*)addr, (unsigned int)__float_as_int(val));
}

// ---------------- KG aggregation: gather-multiply-scatter-mean ----------------
// one wave per edge; lane handles 2 channels (float2) -> 256B coalesced rows

__global__ void k_kg_agg(const float* __restrict__ ent, const float* __restrict__ w,
                         const int* __restrict__ head, const int* __restrict__ tail,
                         const int* __restrict__ etype,
                         float* __restrict__ s, float* __restrict__ cnt) {
  int e = blockIdx.x * (blockDim.x >> 5) + (threadIdx.x >> 5);
  int lane = threadIdx.x & 31;
  if (e >= N_KG_EDGES) return;
  int hd = head[e], tl = tail[e], r = etype[e];
  float2 a = ((const float2*)(ent + (size_t)tl * CH))[lane];
  float2 b = ((const float2*)(w   + (size_t)r  * CH))[lane];
  float* dst = s + (size_t)hd * CH + lane * 2;
  atomicAdd(dst,     a.x * b.x);
  atomicAdd(dst + 1, a.y * b.y);
  if (lane == 0) atomicAdd(cnt + hd, 1.0f);
}

__global__ void k_kg_div(float* __restrict__ s, const float* __restrict__ cnt) {
  size_t i = (size_t)blockIdx.x * blockDim.x + threadIdx.x;
  if (i >= (size_t)N_ENT * CH) return;
  s[i] = s[i] / fmaxf(cnt[i >> 6], 1.0f);
}

// ---------------- lat = softmax(disen_weight_att) @ weight  [F,CH] ----------------

__global__ void k_lat(const float* __restrict__ att, const float* __restrict__ w,
                      float* __restrict__ lat) {
  int t = threadIdx.x;              // 256 threads, 1 block
  int f = t >> 6, c = t & 63;
  float mx = -__builtin_inff();
  for (int r = 0; r < N_REL; r++) mx = fmaxf(mx, att[f * N_REL + r]);
  float sum = 0.0f;
  for (int r = 0; r < N_REL; r++) sum += expf(att[f * N_REL + r] - mx);
  float acc = 0.0f;
  for (int r = 0; r < N_REL; r++)
    acc += (expf(att[f * N_REL + r] - mx) / sum) * w[r * CH + c];
  lat[f * CH + c] = acc;
}

// ---------------- score = softmax(gene_sl_emb @ lat.T, axis=1)  [G,F] ----------------

__global__ void k_score(const float* __restrict__ gemb, const float* __restrict__ lat,
                        float* __restrict__ score) {
  int g = blockIdx.x * blockDim.x + threadIdx.x;
  if (g >= N_GENES) return;
  const float* row = gemb + (size_t)g * CH;
  float d[N_FACTORS];
  for (int f = 0; f < N_FACTORS; f++) {
    float acc = 0.0f;
    for (int c = 0; c < CH; c++) acc += row[c] * lat[f * CH + c];
    d[f] = acc;
  }
  float mx = fmaxf(fmaxf(d[0], d[1]), fmaxf(d[2], d[3]));
  float sum = 0.0f;
  for (int f = 0; f < N_FACTORS; f++) { d[f] = expf(d[f] - mx); sum += d[f]; }
  for (int f = 0; f < N_FACTORS; f++) score[g * N_FACTORS + f] = d[f] / sum;
}

// ---------------- x = entity_agg[reidx]  [G,CH] ----------------

__global__ void k_gather_x(const float* __restrict__ eagg, const int* __restrict__ reidx,
                           float* __restrict__ x) {
  size_t i = (size_t)blockIdx.x * blockDim.x + threadIdx.x;
  if (i >= (size_t)N_GENES * CH) return;
  int g = (int)(i >> 6), c = (int)(i & 63);
  x[i] = eagg[(size_t)reidx[g] * CH + c];
}

// ---------------- gene_agg += vals * entity_emb[reidx[cols]]  (SpMM) ----------------

__global__ void k_spmm(const float* __restrict__ ent, const int* __restrict__ reidx,
                       const float* __restrict__ vals, const int* __restrict__ rows,
                       const int* __restrict__ cols, float* __restrict__ gret) {
  int e = blockIdx.x * (blockDim.x >> 5) + (threadIdx.x >> 5);
  int lane = threadIdx.x & 31;
  if (e >= NNZ) return;
  int rrow = rows[e], ccol = cols[e];
  float v = vals[e];
  int ei = reidx[ccol];
  float2 a = ((const float2*)(ent + (size_t)ei * CH))[lane];
  float* dst = gret + (size_t)rrow * CH + lane * 2;
  atomicAdd(dst,     v * a.x);
  atomicAdd(dst + 1, v * a.y);
}

// ---------------- h[f] = x @ gat_W[f]  via V_WMMA_F32_16X16X4_F32 ----------------
// one wave per 16x16 output tile; 16 WMMAs over K=64.
// A 16x4 layout: lane<16 -> M=lane, K={k0,k0+1}; lane>=16 -> M=lane-16, K={k0+2,k0+3}
// B 4x16 layout: row striped across lanes, same half-wave K offset
// C/D: VGPR j holds row m0 + j + 8*(lane>=16), col n0 + (lane&15)

__global__ void k_gemm_h(const float* __restrict__ x, const float* __restrict__ W,
                         float* __restrict__ h) {
  int mt = blockIdx.x, nt = blockIdx.y, f = blockIdx.z;
  int lane = threadIdx.x;
  int half = lane >> 4;
  int l = lane & 15;
  int m0 = mt * 16, n0 = nt * 16;
  const float* Wf = W + (size_t)f * CH * CH;
  const float* xrow = x + (size_t)(m0 + l) * CH;
  v8f acc = {};
#pragma unroll
  for (int k0 = 0; k0 < CH; k0 += 4) {
    int ka = k0 + half * 2;
    v2f a; a.x = xrow[ka];                 a.y = xrow[ka + 1];
    v2f b; b.x = Wf[ka * CH + n0 + l];     b.y = Wf[(ka + 1) * CH + n0 + l];
    acc = __builtin_amdgcn_wmma_f32_16x16x4_f32(false, a, false, b,
                                                (short)0, acc, false, false);
  }
  float* hf = h + (size_t)f * N_GENES * CH;
#pragma unroll
  for (int j = 0; j < 8; j++)
    hf[(size_t)(m0 + j + half * 8) * CH + n0 + l] = acc[j];
}

// ---------------- el/er = (h * a{l,r}).sum(-1)  [F,G] ----------------

__global__ void k_elr(const float* __restrict__ h, const float* __restrict__ al,
                      const float* __restrict__ ar, float* __restrict__ el,
                      float* __restrict__ er) {
  int id = blockIdx.x * blockDim.x + threadIdx.x;
  if (id >= N_FACTORS * N_GENES) return;
  int f = id / N_GENES, g = id % N_GENES;
  const float* hr = h + ((size_t)f * N_GENES + g) * CH;
  float sl = 0.0f, sr = 0.0f;
  for (int c = 0; c < CH; c++) {
    float hv = hr[c];
    sl += hv * al[f * CH + c];
    sr += hv * ar[f * CH + c];
  }
  el[id] = sl; er[id] = sr;
}

// ---------------- edge softmax (3 passes, recompute e each pass) ----------------

__device__ __forceinline__ float edge_e(const float* el, const float* er,
                                        int f, int s, int d) {
  float ev = el[f * N_GENES + s] + er[f * N_GENES + d];
  return ev > 0.0f ? ev : 0.2f * ev;  // leaky_relu(., 0.2)
}

__global__ void k_edge_max(const int* __restrict__ src, const int* __restrict__ dst,
                           const float* __restrict__ el, const float* __restrict__ er,
                           float* __restrict__ m) {
  int id = blockIdx.x * blockDim.x + threadIdx.x;
  if (id >= N_FACTORS * N_SL_EDGES) return;
  int f = id / N_SL_EDGES, e = id % N_SL_EDGES;
  int s = src[e], d = dst[e];
  atomicMaxF(m + f * N_GENES + d, edge_e(el, er, f, s, d));
}

__global__ void k_edge_z(const int* __restrict__ src, const int* __restrict__ dst,
                         const float* __restrict__ el, const float* __restrict__ er,
                         const float* __restrict__ m, float* __restrict__ z) {
  int id = blockIdx.x * blockDim.x + threadIdx.x;
  if (id >= N_FACTORS * N_SL_EDGES) return;
  int f = id / N_SL_EDGES, e = id % N_SL_EDGES;
  int s = src[e], d = dst[e];
  float ex = expf(edge_e(el, er, f, s, d) - m[f * N_GENES + d]);
  atomicAdd(z + f * N_GENES + d, ex);
}

// one wave per (f,edge); lane handles 2 channels
__global__ void k_edge_aggr(const int* __restrict__ src, const int* __restrict__ dst,
                            const float* __restrict__ el, const float* __restrict__ er,
                            const float* __restrict__ m, const float* __restrict__ z,
                            const float* __restrict__ h, float* __restrict__ outs) {
  int id = blockIdx.x * (blockDim.x >> 5) + (threadIdx.x >> 5);
  int lane = threadIdx.x & 31;
  if (id >= N_FACTORS * N_SL_EDGES) return;
  int f = id / N_SL_EDGES, e = id % N_SL_EDGES;
  int s = src[e], d = dst[e];
  float alpha = expf(edge_e(el, er, f, s, d) - m[f * N_GENES + d]) / z[f * N_GENES + d];
  float2 hv = ((const float2*)(h + ((size_t)f * N_GENES + s) * CH))[lane];
  float* o = outs + ((size_t)f * N_GENES + d) * CH + lane * 2;
  atomicAdd(o,     alpha * hv.x);
  atomicAdd(o + 1, alpha * hv.y);
}

// ---------------- final: output_sl + gene_return ----------------

__global__ void k_final(const float* __restrict__ outs, const float* __restrict__ lat,
                        const float* __restrict__ score, float* __restrict__ gret,
                        float* __restrict__ osl) {
  int i = blockIdx.x * blockDim.x + threadIdx.x;
  if (i >= N_GENES * CH) return;
  int g = i >> 6, c = i & 63;
  float acc = 0.0f;
#pragma unroll
  for (int f = 0; f < N_FACTORS; f++) {
    float v = outs[((size_t)f * N_GENES + g) * CH + c] * lat[f * CH + c];
    osl[(size_t)g * N_FACTORS * CH + f * CH + c] = v;
    acc += v * score[g * N_FACTORS + f];
  }
  gret[i] += acc;
}

// ---------------- launch ----------------

extern "C" void kernel_launch(void* const* d_in, const int* in_sizes, int n_in,
                              void* d_out, int out_size, void* d_ws, size_t ws_size,
                              hipStream_t stream) {
  const float* entity_emb  = (const float*)d_in[0];
  const float* gene_sl_emb = (const float*)d_in[1];
  // d_in[2] latent_emb is overwritten by the reference -> unused
  const float* weight      = (const float*)d_in[3];
  const float* datt        = (const float*)d_in[4];
  const float* ivals       = (const float*)d_in[5];
  const float* gat_W       = (const float*)d_in[6];
  const float* gat_al      = (const float*)d_in[7];
  const float* gat_ar      = (const float*)d_in[8];
  const int*   edge_index  = (const int*)d_in[9];
  const int*   edge_type   = (const int*)d_in[10];
  const int*   reidx       = (const int*)d_in[11];
  const int*   irows       = (const int*)d_in[12];
  const int*   icols       = (const int*)d_in[13];
  const int*   sl_src      = (const int*)d_in[14];
  const int*   sl_dst      = (const int*)d_in[15];

  const int* head = edge_index;
  const int* tail = edge_index + N_KG_EDGES;

  float* entity_agg = (float*)d_out;                       // [N_ENT, CH]
  float* gene_ret   = entity_agg + (size_t)N_ENT * CH;     // [G, CH]
  float* out_sl     = gene_ret + (size_t)N_GENES * CH;     // [G, F, CH]

  float* ws    = (float*)d_ws;
  float* cnt   = ws;                                       // N_ENT
  float* lat   = cnt + N_ENT;                              // F*CH
  float* score = lat + N_FACTORS * CH;                     // F*G
  float* xg    = score + N_FACTORS * N_GENES;              // G*CH
  float* h     = xg + (size_t)N_GENES * CH;                // F*G*CH
  float* el    = h + (size_t)N_FACTORS * N_GENES * CH;     // F*G
  float* er    = el + N_FACTORS * N_GENES;                 // F*G
  float* mbuf  = er + N_FACTORS * N_GENES;                 // F*G
  float* z     = mbuf + N_FACTORS * N_GENES;               // F*G
  float* outs  = z + N_FACTORS * N_GENES;                  // F*G*CH

  // zero accumulators (kernel fills: graph-capture safe)
  {
    size_t n = (size_t)out_size;
    k_zero<<<(unsigned)((n + 255) / 256), 256, 0, stream>>>((float*)d_out, n);
    n = N_ENT;
    k_zero<<<(unsigned)((n + 255) / 256), 256, 0, stream>>>(cnt, n);
    n = (size_t)N_FACTORS * N_GENES;
    k_zero<<<(unsigned)((n + 255) / 256), 256, 0, stream>>>(z, n);
    n = (size_t)N_FACTORS * N_GENES * CH;
    k_zero<<<(unsigned)((n + 255) / 256), 256, 0, stream>>>(outs, n);
  }
  k_init_minf<<<(N_FACTORS * N_GENES + 255) / 256, 256, 0, stream>>>(mbuf, N_FACTORS * N_GENES);

  // KG aggregation (memory-bound hot loop)
  k_kg_agg<<<(N_KG_EDGES + 7) / 8, 256, 0, stream>>>(entity_emb, weight, head, tail,
                                                     edge_type, entity_agg, cnt);
  k_kg_div<<<(unsigned)(((size_t)N_ENT * CH + 255) / 256), 256, 0, stream>>>(entity_agg, cnt);

  // latent / score
  k_lat<<<1, 256, 0, stream>>>(datt, weight, lat);
  k_score<<<(N_GENES + 255) / 256, 256, 0, stream>>>(gene_sl_emb, lat, score);

  // gathers + SpMM
  k_gather_x<<<(unsigned)(((size_t)N_GENES * CH + 255) / 256), 256, 0, stream>>>(entity_agg, reidx, xg);
  k_spmm<<<(NNZ + 7) / 8, 256, 0, stream>>>(entity_emb, reidx, ivals, irows, icols, gene_ret);

  // WMMA GEMM: h[f] = x @ gat_W[f]
  {
    dim3 g(N_GENES / 16, CH / 16, N_FACTORS);
    k_gemm_h<<<g, 32, 0, stream>>>(xg, gat_W, h);
  }

  // GAT edge softmax + aggregation
  k_elr<<<(N_FACTORS * N_GENES + 255) / 256, 256, 0, stream>>>(h, gat_al, gat_ar, el, er);
  k_edge_max<<<(N_FACTORS * N_SL_EDGES + 255) / 256, 256, 0, stream>>>(sl_src, sl_dst, el, er, mbuf);
  k_edge_z<<<(N_FACTORS * N_SL_EDGES + 255) / 256, 256, 0, stream>>>(sl_src, sl_dst, el, er, mbuf, z);
  k_edge_aggr<<<(N_FACTORS * N_SL_EDGES + 7) / 8, 256, 0, stream>>>(sl_src, sl_dst, el, er,
                                                                    mbuf, z, h, outs);

  // final combine
  k_final<<<(N_GENES * CH + 255) / 256, 256, 0, stream>>>(outs, lat, score, gene_ret, out_sl);
}